// SimilarityRegistration_15006615733259
// MI455X (gfx1250) — compile-verified
//
#include <hip/hip_runtime.h>
#include <hip/hip_bf16.h>

// Problem constants (match reference)
#define HWDIM 512
#define NPIX  (HWDIM * HWDIM)      // 262144
#define NITER 100
#define TPB   256
#define PPT   4                    // consecutive pixels per thread
#define NBLK  (NPIX / (TPB * PPT)) // 256 blocks (== TPB, one partial per thread in finisher)

typedef float v2f __attribute__((ext_vector_type(2)));
typedef float v8f __attribute__((ext_vector_type(8)));

// ---------------------------------------------------------------------------
// Bilinear corner gather with zeros padding (matches jnp.where/clip semantics:
// clip -> int truncation for the index, validity mask on the *unclipped* float
// coordinate).
// ---------------------------------------------------------------------------
static __device__ __forceinline__ float corner_gather(const float* __restrict__ img,
                                                      float xf, float yf) {
    bool valid = (xf >= 0.0f) && (xf <= (float)(HWDIM - 1)) &&
                 (yf >= 0.0f) && (yf <= (float)(HWDIM - 1));
    int xi = (int)fminf(fmaxf(xf, 0.0f), (float)(HWDIM - 1));
    int yi = (int)fminf(fmaxf(yf, 0.0f), (float)(HWDIM - 1));
    float v = img[(yi << 9) | xi];
    return valid ? v : 0.0f;
}

// ---------------------------------------------------------------------------
// Exact-f32 wave32 reduction of TWO lane-private scalars via one
// V_WMMA_F32_16X16X4_F32 (A = {a0,a1} striped 16x4; B selects column 0 for a0
// and column 1 for a1 => bv = 1.0 iff lane in {0,17}). Column sums land in
// lanes {0,16} / {1,17}; summing the 8 D VGPRs in-lane + 2 shuffles finishes.
// Requires EXEC all-ones (call from uniform control flow).
// ---------------------------------------------------------------------------
static __device__ __forceinline__ void wave_reduce2_wmma(float a0, float a1,
                                                         float& r0, float& r1) {
    const int lane = threadIdx.x & 31;
    const float bv = (lane == 0 || lane == 17) ? 1.0f : 0.0f;
    v2f A; A.x = a0; A.y = a1;
    v2f B; B.x = bv; B.y = bv;
    v8f C = {};
    v8f D = __builtin_amdgcn_wmma_f32_16x16x4_f32(
        /*neg_a=*/false, A, /*neg_b=*/false, B,
        /*c_mod=*/(short)0, C, /*reuse_a=*/false, /*reuse_b=*/false);
    float t = ((D[0] + D[1]) + (D[2] + D[3])) + ((D[4] + D[5]) + (D[6] + D[7]));
    r0 = __shfl(t, 0) + __shfl(t, 16);
    r1 = __shfl(t, 1) + __shfl(t, 17);
}

// ---------------------------------------------------------------------------
// ws layout: float ws[0..3] = params {s, r, tx, ty}
//            int   ((int*)ws)[4] = arrival counter (reset by finisher block)
//            float ws[8 + 4*b + c] = per-block gradient partials (b<NBLK, c<4)
// ---------------------------------------------------------------------------

__global__ void k_init(const float* __restrict__ scale,
                       const float* __restrict__ rot,
                       const float* __restrict__ trans,
                       float* __restrict__ ws) {
    if (threadIdx.x == 0) {
        ws[0] = scale[0];
        ws[1] = rot[0];
        ws[2] = trans[0];
        ws[3] = trans[1];
        ((int*)ws)[4] = 0;
    }
}

// One pass: compute 4-scalar gradient of MSE w.r.t. {s,r,tx,ty}, and the
// last-arriving block applies the SGD update (LR=1) in-kernel.
__global__ __launch_bounds__(TPB) void k_grad(const float* __restrict__ mov,
                                              const float* __restrict__ fix,
                                              float* __restrict__ ws) {
    const float s  = ws[0];
    const float r  = ws[1];
    const float tx = ws[2];
    const float ty = ws[3];
    const float c  = cosf(r);
    const float sn = sinf(r);
    const float scv = s * c;
    const float ssv = s * sn;

    // This thread owns PPT consecutive pixels of one row.
    const int idx0 = (blockIdx.x * TPB + threadIdx.x) * PPT;
    const int i  = idx0 >> 9;
    const int j0 = idx0 & (HWDIM - 1);
    const float ys = ((float)i + 0.5f) * (2.0f / (float)HWDIM) - 1.0f;
    const float gxc = -ssv * ys + tx;   // row-constant parts
    const float gyc =  scv * ys + ty;

    const float4 f4 = *(const float4*)(fix + idx0);  // one B128 load
    const float fv[PPT] = {f4.x, f4.y, f4.z, f4.w};

    float gs = 0.0f, gr = 0.0f, gtx = 0.0f, gty = 0.0f;

#pragma unroll
    for (int k = 0; k < PPT; ++k) {
        const float xs = ((float)(j0 + k) + 0.5f) * (2.0f / (float)HWDIM) - 1.0f;
        const float gx = scv * xs + gxc;
        const float gy = ssv * xs + gyc;
        // unnormalize (align_corners=False): ix = ((gx+1)*W - 1)*0.5
        const float ix = gx * 256.0f + 255.5f;
        const float iy = gy * 256.0f + 255.5f;

        const float x0 = floorf(ix);
        const float y0 = floorf(iy);
        const float wx1 = ix - x0, wx0 = 1.0f - wx1;
        const float wy1 = iy - y0, wy0 = 1.0f - wy1;

        const float v00 = corner_gather(mov, x0,        y0);
        const float v01 = corner_gather(mov, x0 + 1.0f, y0);
        const float v10 = corner_gather(mov, x0,        y0 + 1.0f);
        const float v11 = corner_gather(mov, x0 + 1.0f, y0 + 1.0f);

        const float tr = wy0 * (wx0 * v00 + wx1 * v01) + wy1 * (wx0 * v10 + wx1 * v11);
        const float g  = (tr - fv[k]) * (2.0f / (float)NPIX);

        // Gradient flows only through the bilinear weights (JAX semantics).
        const float dtdix = wy0 * (v01 - v00) + wy1 * (v11 - v10);
        const float dtdiy = wx0 * (v10 - v00) + wx1 * (v11 - v01);
        const float dgx = g * dtdix * 256.0f;  // dL/d gx
        const float dgy = g * dtdiy * 256.0f;  // dL/d gy

        gs  += dgx * (c * xs - sn * ys)     + dgy * (sn * xs + c * ys);
        gr  += dgx * (-ssv * xs - scv * ys) + dgy * (scv * xs - ssv * ys);
        gtx += dgx;
        gty += dgy;
    }

    // ---- block reduction: WMMA per wave, then LDS across the 8 waves ----
    __shared__ float red[TPB / 32][4];
    __shared__ int amLast;
    const int lane = threadIdx.x & 31;
    const int wid  = threadIdx.x >> 5;

    float ra, rb, rc, rd;
    wave_reduce2_wmma(gs,  gr,  ra, rb);
    wave_reduce2_wmma(gtx, gty, rc, rd);
    if (lane == 0) {
        red[wid][0] = ra; red[wid][1] = rb; red[wid][2] = rc; red[wid][3] = rd;
    }
    __syncthreads();
    if (threadIdx.x < 4) {
        float acc = 0.0f;
#pragma unroll
        for (int w = 0; w < TPB / 32; ++w) acc += red[w][threadIdx.x];
        ws[8 + blockIdx.x * 4 + threadIdx.x] = acc;  // deterministic partial
    }

    // ---- fused finalize: last-arriving block applies the SGD step ----
    __threadfence();  // release partials before signaling arrival
    if (threadIdx.x == 0) {
        int prev = atomicAdd((int*)ws + 4, 1);
        amLast = (prev == NBLK - 1) ? 1 : 0;
    }
    __syncthreads();          // amLast is block-uniform; no wave divergence below
    if (amLast) {
        __threadfence();      // acquire all blocks' partials
        const float4 p = *((const float4*)(ws + 8) + threadIdx.x);  // NBLK==TPB
        float q0, q1, q2, q3;
        wave_reduce2_wmma(p.x, p.y, q0, q1);
        wave_reduce2_wmma(p.z, p.w, q2, q3);
        if (lane == 0) {
            red[wid][0] = q0; red[wid][1] = q1; red[wid][2] = q2; red[wid][3] = q3;
        }
        __syncthreads();
        if (threadIdx.x == 0) {
            float g0 = 0.0f, g1 = 0.0f, g2 = 0.0f, g3 = 0.0f;
#pragma unroll
            for (int w = 0; w < TPB / 32; ++w) {
                g0 += red[w][0]; g1 += red[w][1]; g2 += red[w][2]; g3 += red[w][3];
            }
            ws[0] = s  - g0;   // LR = 1.0
            ws[1] = r  - g1;
            ws[2] = tx - g2;
            ws[3] = ty - g3;
            ((int*)ws)[4] = 0; // reset arrival counter for next pass / replay
        }
    }
}

__global__ __launch_bounds__(TPB) void k_final(const float* __restrict__ mov,
                                               const float* __restrict__ ws,
                                               float* __restrict__ out) {
    const float s  = ws[0];
    const float r  = ws[1];
    const float tx = ws[2];
    const float ty = ws[3];
    const float c  = cosf(r);
    const float sn = sinf(r);
    const float scv = s * c;
    const float ssv = s * sn;

    const int idx0 = (blockIdx.x * TPB + threadIdx.x) * PPT;
    const int i  = idx0 >> 9;
    const int j0 = idx0 & (HWDIM - 1);
    const float ys = ((float)i + 0.5f) * (2.0f / (float)HWDIM) - 1.0f;
    const float gxc = -ssv * ys + tx;
    const float gyc =  scv * ys + ty;

    float4 o;
    float* ov = (float*)&o;
#pragma unroll
    for (int k = 0; k < PPT; ++k) {
        const float xs = ((float)(j0 + k) + 0.5f) * (2.0f / (float)HWDIM) - 1.0f;
        const float gx = scv * xs + gxc;
        const float gy = ssv * xs + gyc;
        const float ix = gx * 256.0f + 255.5f;
        const float iy = gy * 256.0f + 255.5f;

        const float x0 = floorf(ix);
        const float y0 = floorf(iy);
        const float wx1 = ix - x0, wx0 = 1.0f - wx1;
        const float wy1 = iy - y0, wy0 = 1.0f - wy1;

        const float v00 = corner_gather(mov, x0,        y0);
        const float v01 = corner_gather(mov, x0 + 1.0f, y0);
        const float v10 = corner_gather(mov, x0,        y0 + 1.0f);
        const float v11 = corner_gather(mov, x0 + 1.0f, y0 + 1.0f);

        ov[k] = wy0 * (wx0 * v00 + wx1 * v01) + wy1 * (wx0 * v10 + wx1 * v11);
    }
    *(float4*)(out + idx0) = o;  // one B128 store
}

extern "C" void kernel_launch(void* const* d_in, const int* in_sizes, int n_in,
                              void* d_out, int out_size, void* d_ws, size_t ws_size,
                              hipStream_t stream) {
    const float* mov   = (const float*)d_in[0];  // [1,512,512] f32
    const float* fix   = (const float*)d_in[1];  // [1,512,512] f32
    const float* scale = (const float*)d_in[2];  // [1] f32
    const float* rot   = (const float*)d_in[3];  // [1] f32
    const float* trans = (const float*)d_in[4];  // [2] f32
    float* ws  = (float*)d_ws;
    float* out = (float*)d_out;

    k_init<<<1, 32, 0, stream>>>(scale, rot, trans, ws);
    for (int it = 0; it < NITER; ++it) {
        k_grad<<<NBLK, TPB, 0, stream>>>(mov, fix, ws);  // update fused in-kernel
    }
    k_final<<<NBLK, TPB, 0, stream>>>(mov, ws, out);
}